// DistMult_30090540876231
// MI455X (gfx1250) — compile-verified
//
#include <hip/hip_runtime.h>
#include <hip/hip_bf16.h>
#include <stdint.h>

#define EMB 128

typedef float        v2f   __attribute__((ext_vector_type(2)));
typedef float        v8f   __attribute__((ext_vector_type(8)));
typedef unsigned int u32x4 __attribute__((ext_vector_type(4)));
typedef int          i32x4 __attribute__((ext_vector_type(4)));
typedef int          i32x8 __attribute__((ext_vector_type(8)));

// ---------------------------------------------------------------------------
// TDM: DMA fwd_rel (numRels x 128 f32, contiguous) from global into LDS @ 0.
// Descriptor bitfields per CDNA5 ISA ch.8 (async/tensor doc).
// ---------------------------------------------------------------------------
__device__ __forceinline__ void tdm_load_rel_to_lds(const float* rel, int numRels) {
#if __has_builtin(__builtin_amdgcn_tensor_load_to_lds)
  unsigned long long ga = (unsigned long long)(uintptr_t)rel;

  u32x4 g0;
  g0.x = 1u;                                              // count=1, user D#, no gather
  g0.y = 0u;                                              // lds_addr = 0 (dynamic LDS base)
  g0.z = (unsigned)(ga & 0xFFFFFFFFull);                  // global_addr[31:0]
  g0.w = (unsigned)((ga >> 32) & 0x01FFFFFFull) | (2u << 30); // global_addr[56:32] | type=2

  const unsigned td0 = EMB;                 // tensor_dim0 (elements)
  const unsigned td1 = (unsigned)numRels;   // tensor_dim1
  const unsigned tl0 = EMB;                 // tile_dim0
  const unsigned tl1 = (unsigned)numRels;   // tile_dim1 (whole table = one tile)
  const unsigned long long s0 = EMB;                              // dim0 stride
  const unsigned long long s1 = (unsigned long long)EMB * td1;    // dim1 stride

  i32x8 g1;
  g1[0] = (int)(2u << 16);                                          // data_size = 4B
  g1[1] = (int)((td0 & 0xFFFFu) << 16);                             // tensor_dim0[15:0]
  g1[2] = (int)(((td0 >> 16) & 0xFFFFu) | ((td1 & 0xFFFFu) << 16)); // dim0 hi | dim1 lo
  g1[3] = (int)(((td1 >> 16) & 0xFFFFu) | ((tl0 & 0xFFFFu) << 16)); // dim1 hi | tile0
  g1[4] = (int)(tl1 & 0xFFFFu);                                     // tile1 | tile2=0
  g1[5] = (int)(unsigned)(s0 & 0xFFFFFFFFull);                      // stride0[31:0]
  g1[6] = (int)((unsigned)((s0 >> 32) & 0xFFFFull) |
                ((unsigned)(s1 & 0xFFFFull) << 16));                // stride0 hi | stride1 lo
  g1[7] = (int)(unsigned)((s1 >> 16) & 0xFFFFFFFFull);              // stride1[47:16]

  i32x4 gz = {0, 0, 0, 0};
#if defined(__clang_major__) && (__clang_major__ >= 23)
  i32x8 gz8 = {0, 0, 0, 0, 0, 0, 0, 0};
  __builtin_amdgcn_tensor_load_to_lds(g0, g1, gz, gz, gz8, 0);
#else
  __builtin_amdgcn_tensor_load_to_lds(g0, g1, gz, gz, 0);
#endif
#endif
}

// ---------------------------------------------------------------------------
// DistMult scoring: one wave32 handles a tile of 16 edges.
// A[m][k] = (h[src]⊙w)[m][4kc+k], B[k][n] = h[dst][n][4kc+k];
// diagonal of D accumulates the 16 dot products via v_wmma_f32_16x16x4_f32.
// Lane l serves edge (l&15); lanes 0-15 carry K={0,1}, lanes 16-31 K={2,3},
// folded into the row pointers as a +2-float offset so every inner-loop
// access is a branch-free b64 load with an immediate offset.
// ---------------------------------------------------------------------------
__global__ __launch_bounds__(1024, 1)
void distmult_wmma_kernel(const float* __restrict__ h,
                          const int*   __restrict__ srcI,
                          const int*   __restrict__ dstI,
                          const int*   __restrict__ etI,
                          const float* __restrict__ rel,
                          float*       __restrict__ out,
                          int E, int numRels)
{
  extern __shared__ float s_rel[];   // numRels * 128 floats (<= 256 KB)

#if __has_builtin(__builtin_amdgcn_tensor_load_to_lds)
  if (threadIdx.x < 32) {            // wave 0 issues the DMA (EXEC ignored by TDM)
    tdm_load_rel_to_lds(rel, numRels);
  }
  __builtin_amdgcn_s_wait_tensorcnt(0);  // no-op for non-issuing waves
#else
  for (int i = (int)threadIdx.x; i < numRels * EMB; i += (int)blockDim.x)
    s_rel[i] = rel[i];
#endif
  __syncthreads();

  const int lane          = (int)(threadIdx.x & 31u);
  const int waveInBlock   = (int)(threadIdx.x >> 5);
  const int wavesPerBlock = (int)(blockDim.x >> 5);
  const int waveId        = (int)blockIdx.x * wavesPerBlock + waveInBlock;
  const int nWaves        = (int)gridDim.x * wavesPerBlock;

  const int m    = lane & 15;              // edge-in-tile this lane serves
  const int sel2 = (lane >= 16) ? 2 : 0;   // K-pair offset (floats) for this lane

  const int nTiles = (E + 15) >> 4;

  for (int tile = waveId; tile < nTiles; tile += nWaves) {
    const int ebase = tile << 4;
    int e = ebase + m;
    if (e >= E) e = E - 1;           // tail clamp (loads only; stores guarded below)

    const float* __restrict__ hu = h     + (size_t)srcI[e] * EMB + sel2;
    const float* __restrict__ hv = h     + (size_t)dstI[e] * EMB + sel2;
    const float* __restrict__ wp = s_rel + (size_t)etI[e]  * EMB + sel2;

    v8f acc = {0.f, 0.f, 0.f, 0.f, 0.f, 0.f, 0.f, 0.f};

#pragma unroll 8
    for (int kc = 0; kc < EMB / 4; ++kc) {
      const float2 u2 = *reinterpret_cast<const float2*>(hu + (kc << 2)); // global b64
      const float2 v2 = *reinterpret_cast<const float2*>(hv + (kc << 2)); // global b64
      const float2 w2 = *reinterpret_cast<const float2*>(wp + (kc << 2)); // ds b64

      v2f A, B;
      A.x = u2.x * w2.x;   // A vgpr0: K=0 (lanes 0-15) / K=2 (lanes 16-31)
      A.y = u2.y * w2.y;   // A vgpr1: K=1 / K=3
      B.x = v2.x;          // B vgpr0: K=0 / K=2, N = lane&15
      B.y = v2.y;          // B vgpr1: K=1 / K=3

      acc = __builtin_amdgcn_wmma_f32_16x16x4_f32(
          /*neg_a=*/false, A, /*neg_b=*/false, B,
          /*c_mod=*/(short)0, acc, /*reuse_a=*/false, /*reuse_b=*/false);
    }

    // Diagonal D[m][m]: m<8 -> lane m, c[m];  m>=8 -> lane m+16, c[m-8]
    const float diag   = acc[lane & 7];
    const bool  writer = (lane < 8) || (lane >= 24);
    const int   eo     = ebase + ((lane < 8) ? lane : (lane - 16));
    if (writer && eo < E) out[eo] = diag;
  }
}

extern "C" void kernel_launch(void* const* d_in, const int* in_sizes, int n_in,
                              void* d_out, int out_size, void* d_ws, size_t ws_size,
                              hipStream_t stream) {
  const float* h    = (const float*)d_in[0];
  const int*   srcI = (const int*)  d_in[1];
  const int*   dstI = (const int*)  d_in[2];
  const int*   etI  = (const int*)  d_in[3];
  const float* rel  = (const float*)d_in[4];
  float*       out  = (float*)d_out;

  const int E       = in_sizes[1];
  const int numRels = in_sizes[4] / EMB;
  const size_t ldsBytes = (size_t)numRels * EMB * sizeof(float);

  (void)hipFuncSetAttribute((const void*)distmult_wmma_kernel,
                            hipFuncAttributeMaxDynamicSharedMemorySize,
                            (int)ldsBytes);

  const int threads       = 1024;                 // 32 wave32s, 1 WG/WGP (LDS-limited)
  const int wavesPerBlock = threads / 32;
  const int nTiles        = (E + 15) / 16;
  int blocks = (nTiles + wavesPerBlock - 1) / wavesPerBlock;
  if (blocks > 512) blocks = 512;
  if (blocks < 1)   blocks = 1;

  distmult_wmma_kernel<<<blocks, threads, ldsBytes, stream>>>(
      h, srcI, dstI, etI, rel, out, E, numRels);
}